// MultiReDiffusion_18107582120670
// MI455X (gfx1250) — compile-verified
//
#include <hip/hip_runtime.h>

// Problem constants from the reference
#define RDIM 4
#define SDIM 8
#define NDIM 4096
#define DDIM 64

typedef __attribute__((ext_vector_type(2))) float v2f;
typedef __attribute__((ext_vector_type(8))) float v8f;

// D = A(16x4 f32) * B(4x16 f32) + C(16x16 f32), full-precision matrix op.
// Arg pattern follows the probed 8-arg 16x16xK family:
// (neg_a, A, neg_b, B, c_mod, C, reuse_a, reuse_b)
__device__ __forceinline__ v8f wmma_f32_4(v2f a, v2f b, v8f c) {
  return __builtin_amdgcn_wmma_f32_16x16x4_f32(false, a, false, b, (short)0, c,
                                               false, false);
}

// ---------------------------------------------------------------------------
// Kernel 0: Y[r] = X[r] @ fc_w[r]^T   (folded FC, tiny GEMM, K=64)
// One wave -> 16 rows x 64 cols. Block = 4 waves -> 64 rows.
// ---------------------------------------------------------------------------
__global__ __launch_bounds__(128) void mrd_xw_kernel(
    const float* __restrict__ x, const float* __restrict__ fc_w,
    float* __restrict__ y) {
  const int lane = threadIdx.x & 31;
  const int wv = threadIdx.x >> 5;
  const int lm = lane & 15;   // row (A) / col (B,D) within tile
  const int kh = lane >> 4;   // half selector: K offset 0 or 2
  const int r = blockIdx.y;
  const int row0 = blockIdx.x * 64 + wv * 16;

  const float* xr = x + ((size_t)r * NDIM + row0) * DDIM;
  const float* wr = fc_w + (size_t)r * DDIM * DDIM;  // [o][d]

  v8f acc[4] = {};
#pragma unroll
  for (int k0 = 0; k0 < DDIM; k0 += 4) {
    // A fragment: row = lm, K = k0 + 2*kh + {0,1}
    v2f af = *(const v2f*)(xr + (size_t)lm * DDIM + k0 + 2 * kh);
#pragma unroll
    for (int ni = 0; ni < 4; ++ni) {
      // B fragment: B[k][o] = fc_w[o][k], o = ni*16+lm, consecutive k -> b64
      v2f bf = *(const v2f*)(wr + (size_t)(ni * 16 + lm) * DDIM + k0 + 2 * kh);
      acc[ni] = wmma_f32_4(af, bf, acc[ni]);
    }
  }

  float* yr = y + ((size_t)r * NDIM + row0) * DDIM;
#pragma unroll
  for (int ni = 0; ni < 4; ++ni) {
#pragma unroll
    for (int v = 0; v < 8; ++v) {
      // D layout: row = v + 8*kh, col = ni*16 + lm
      yr[(size_t)(v + 8 * kh) * DDIM + ni * 16 + lm] = acc[ni][v];
    }
  }
}

// ---------------------------------------------------------------------------
// Kernel 1: diffusions = prelu(coef[r] * (A[r] @ Y[r]) + fc_b[r], alpha0)
// The HBM-bound streaming GEMM: 268 MB of A read exactly once.
// One wave -> 16 rows x full 64-col output (4 WMMA tiles), K = 4096.
// Block = 4 waves (64 rows); grid = (N/64, R) = 256 workgroups.
// ---------------------------------------------------------------------------
__global__ __launch_bounds__(128) void mrd_diffusion_kernel(
    const float* __restrict__ theta, const float* __restrict__ t,
    const float* __restrict__ a, const float* __restrict__ y,
    const float* __restrict__ fc_b, const float* __restrict__ alpha0p,
    float* __restrict__ diff) {
  const int lane = threadIdx.x & 31;
  const int wv = threadIdx.x >> 5;
  const int lm = lane & 15;
  const int kh = lane >> 4;
  const int r = blockIdx.y;
  const int row0 = blockIdx.x * 64 + wv * 16;

  // coef[r] = sum_s theta[r,s] * t[r,s]  (uniform -> scalar loads)
  float coef = 0.f;
#pragma unroll
  for (int s = 0; s < SDIM; ++s) coef += theta[r * SDIM + s] * t[r * SDIM + s];
  const float alpha0 = alpha0p[0];

  // Per-lane A stream: row = row0+lm, columns 2*kh + k0 + {0,1}
  const float* arow =
      a + (size_t)r * NDIM * NDIM + (size_t)(row0 + lm) * NDIM + 2 * kh;
  // Y fragment base: Y[r][k0 + 2*kh + {0,1}][ni*16 + lm]
  const float* yb =
      y + (size_t)r * NDIM * DDIM + (size_t)(2 * kh) * DDIM + lm;

  v8f acc[4] = {};
#pragma unroll 4
  for (int k0 = 0; k0 < NDIM; k0 += 4) {
    v2f af = *(const v2f*)(arow + k0);  // 8B/lane, line-complete over 8 iters
    const float* yp = yb + (size_t)k0 * DDIM;
#pragma unroll
    for (int ni = 0; ni < 4; ++ni) {
      v2f bf;
      bf.x = yp[ni * 16];         // K = k0 + 2*kh
      bf.y = yp[ni * 16 + DDIM];  // K = k0 + 2*kh + 1
      acc[ni] = wmma_f32_4(af, bf, acc[ni]);
    }
  }

  // Epilogue straight from D-register layout: scale, bias, PReLU, store.
  float* dr = diff + (size_t)r * NDIM * DDIM + (size_t)row0 * DDIM;
#pragma unroll
  for (int ni = 0; ni < 4; ++ni) {
    const int o = ni * 16 + lm;
    const float bias = fc_b[r * DDIM + o];
#pragma unroll
    for (int v = 0; v < 8; ++v) {
      float val = coef * acc[ni][v] + bias;
      val = (val >= 0.f) ? val : alpha0 * val;
      dr[(size_t)(v + 8 * kh) * DDIM + o] = val;
    }
  }
}

// ---------------------------------------------------------------------------
// Kernel 2: latent[s] = prelu(sum_r conv_w[s,r]*diff[r] + conv_b[s], alpha1)
// Pure elementwise over N*D with a 4-deep channel reduce; coalesced.
// ---------------------------------------------------------------------------
__global__ __launch_bounds__(256) void mrd_conv_kernel(
    const float* __restrict__ diff, const float* __restrict__ conv_w,
    const float* __restrict__ conv_b, const float* __restrict__ alpha1p,
    float* __restrict__ latent) {
  const size_t plane = (size_t)NDIM * DDIM;
  const size_t idx = (size_t)blockIdx.x * blockDim.x + threadIdx.x;
  if (idx >= plane) return;
  const float d0 = diff[idx];
  const float d1 = diff[plane + idx];
  const float d2 = diff[2 * plane + idx];
  const float d3 = diff[3 * plane + idx];
  const float alpha1 = alpha1p[0];
#pragma unroll
  for (int s = 0; s < RDIM; ++s) {
    float v = conv_w[s * RDIM + 0] * d0 + conv_w[s * RDIM + 1] * d1 +
              conv_w[s * RDIM + 2] * d2 + conv_w[s * RDIM + 3] * d3 +
              conv_b[s];
    v = (v >= 0.f) ? v : alpha1 * v;
    latent[s * plane + idx] = v;
  }
}

extern "C" void kernel_launch(void* const* d_in, const int* in_sizes, int n_in,
                              void* d_out, int out_size, void* d_ws,
                              size_t ws_size, hipStream_t stream) {
  (void)in_sizes;
  (void)n_in;
  (void)out_size;
  const float* theta = (const float*)d_in[0];
  const float* t = (const float*)d_in[1];
  const float* a = (const float*)d_in[2];
  const float* x = (const float*)d_in[3];
  const float* fc_w = (const float*)d_in[4];
  const float* fc_b = (const float*)d_in[5];
  const float* conv_w = (const float*)d_in[6];
  const float* conv_b = (const float*)d_in[7];
  const float* alpha0 = (const float*)d_in[8];
  const float* alpha1 = (const float*)d_in[9];

  const size_t plane = (size_t)NDIM * DDIM;  // 262144
  float* out = (float*)d_out;
  float* latent = out;                 // [R, N, D] first tuple element
  float* diff = out + RDIM * plane;    // [R, N, D] second tuple element

  // Scratch for Y = X @ W^T (4 MB). Prefer d_ws; the latent region is free
  // until the final kernel, so it is a valid deterministic fallback.
  float* y = (ws_size >= (size_t)RDIM * plane * sizeof(float)) ? (float*)d_ws
                                                               : latent;

  dim3 g0(NDIM / 64, RDIM);
  mrd_xw_kernel<<<g0, 128, 0, stream>>>(x, fc_w, y);

  dim3 g1(NDIM / 64, RDIM);
  mrd_diffusion_kernel<<<g1, 128, 0, stream>>>(theta, t, a, y, fc_b, alpha0,
                                               diff);

  dim3 g2((unsigned)(plane / 256));
  mrd_conv_kernel<<<g2, 256, 0, stream>>>(diff, conv_w, conv_b, alpha1,
                                          latent);
}